// radon_backward_42984032698901
// MI455X (gfx1250) — compile-verified
//
#include <hip/hip_runtime.h>

#define NANG 720
#define NDET 512
#define KPTS (NANG * NDET)
#define GSZ 1024
#define ALPHA_KB 14.04f       /* 2.34 * J, J = 6 */
#define PI_F 3.14159265358979f
#define NITER_DCOMP 10

typedef float v2f __attribute__((ext_vector_type(2)));
typedef float v8f __attribute__((ext_vector_type(8)));

// ---------------------------------------------------------------------------
// Modified Bessel I0 (Abramowitz & Stegun 9.8.1 / 9.8.2), f32.
// ---------------------------------------------------------------------------
__device__ __forceinline__ float kb_i0f(float x) {
    float ax = fabsf(x);
    if (ax < 3.75f) {
        float t = x / 3.75f;
        t *= t;
        return 1.0f + t * (3.5156229f + t * (3.0899424f + t * (1.2067492f +
                     t * (0.2659732f + t * (0.0360768f + t * 0.0045813f)))));
    } else {
        float t = 3.75f / ax;
        float p = 0.39894228f + t * (0.01328592f + t * (0.00225319f +
                  t * (-0.00157565f + t * (0.00916281f + t * (-0.02057706f +
                  t * (0.02635537f + t * (-0.01647633f + t * 0.00392377f)))))));
        return p * expf(ax) * rsqrtf(ax);
    }
}

// Kaiser-Bessel 1D weights+indices for one coordinate (J=6 taps).
__device__ __forceinline__ void kb6(float om, float wv[6], int idx[6], float inv_i0a) {
    float gx = om * ((float)GSZ / (2.0f * PI_F));
    int k0 = (int)floorf(gx);
#pragma unroll
    for (int j = 0; j < 6; ++j) {
        int id = k0 + (j - 2);               // offs = [-2..3]
        float d = gx - (float)id;
        float r = (2.0f * d) / 6.0f;
        float t = 1.0f - r * r;
        float w = 0.0f;
        if (t > 0.0f) w = kb_i0f(ALPHA_KB * sqrtf(fmaxf(t, 1e-20f)));
        wv[j] = w * inv_i0a;
        idx[j] = id & (GSZ - 1);             // two's-complement & == mod for pow2
    }
}

// ---------------------------------------------------------------------------
// Stage 1: per-angle 512-pt FFT with folded ifftshift/fftshift (checkerboard).
// Stockham radix-2 in LDS, 256 threads (8 wave32) per row.
// ---------------------------------------------------------------------------
__global__ __launch_bounds__(256) void fft512_rows(const float* __restrict__ sino,
                                                   float* __restrict__ kd_r,
                                                   float* __restrict__ kd_i) {
    __shared__ float2 bufA[512];
    __shared__ float2 bufB[512];
    const int row = blockIdx.x;
    const int t = threadIdx.x;
    const float* src = sino + row * NDET;
    for (int e = t; e < 512; e += 256) {
        float sgn = (e & 1) ? -1.0f : 1.0f;  // ifftshift fold
        bufA[e] = make_float2(src[e] * sgn, 0.0f);
    }
    __syncthreads();
    float2* x = bufA;
    float2* y = bufB;
    int l = 256, s = 1;
#pragma unroll
    for (int stage = 0; stage < 9; ++stage) {
        int k = t & (s - 1);
        int j = t >> stage;                  // t / s
        float ang = -PI_F * (float)j / (float)l;
        float sn, cs;
        __sincosf(ang, &sn, &cs);
        float2 a = x[k + s * j];
        float2 b = x[k + s * (j + l)];
        float2 dif = make_float2(a.x - b.x, a.y - b.y);
        y[k + s * (2 * j)] = make_float2(a.x + b.x, a.y + b.y);
        y[k + s * (2 * j + 1)] = make_float2(dif.x * cs - dif.y * sn,
                                             dif.x * sn + dif.y * cs);
        __syncthreads();
        float2* tmp = x; x = y; y = tmp;
        l >>= 1; s <<= 1;
    }
    for (int e = t; e < 512; e += 256) {
        float sgn = (e & 1) ? -1.0f : 1.0f;  // fftshift fold
        kd_r[row * NDET + e] = x[e].x * sgn;
        kd_i[row * NDET + e] = x[e].y * sgn;
    }
}

// ---------------------------------------------------------------------------
// Utility kernels.
// ---------------------------------------------------------------------------
__global__ void zero_f32(float* p, int n) {
    int i = blockIdx.x * blockDim.x + threadIdx.x;
    if (i < n) p[i] = 0.0f;
}
__global__ void fill_f32(float* p, float v, int n) {
    int i = blockIdx.x * blockDim.x + threadIdx.x;
    if (i < n) p[i] = v;
}

// ---------------------------------------------------------------------------
// Density compensation (all real): scatter w, gather |AA^H w|, update w.
// ---------------------------------------------------------------------------
__global__ __launch_bounds__(128) void scatter_real(const float* __restrict__ ksp,
                                                    const float* __restrict__ w,
                                                    float* __restrict__ grid_r) {
    int k = blockIdx.x * blockDim.x + threadIdx.x;
    if (k >= KPTS) return;
    float inv_i0a = 1.0f / kb_i0f(ALPHA_KB);
    float wy[6]; int iy[6];
    float wx[6]; int ix[6];
    kb6(ksp[k], wy, iy, inv_i0a);            // ktraj[0] -> y
    kb6(ksp[KPTS + k], wx, ix, inv_i0a);     // ktraj[1] -> x
    float wk = w[k];
#pragma unroll
    for (int jy = 0; jy < 6; ++jy) {
        float wyk = wk * wy[jy];
        int base = iy[jy] << 10;
#pragma unroll
        for (int jx = 0; jx < 6; ++jx)
            atomicAdd(&grid_r[base + ix[jx]], wyk * wx[jx]);
    }
}

__global__ __launch_bounds__(128) void gather_update(const float* __restrict__ ksp,
                                                     const float* __restrict__ grid_r,
                                                     float* __restrict__ w) {
    int k = blockIdx.x * blockDim.x + threadIdx.x;
    if (k >= KPTS) return;
    float inv_i0a = 1.0f / kb_i0f(ALPHA_KB);
    float wy[6]; int iy[6];
    float wx[6]; int ix[6];
    kb6(ksp[k], wy, iy, inv_i0a);
    kb6(ksp[KPTS + k], wx, ix, inv_i0a);
    float v = 0.0f;
#pragma unroll
    for (int jy = 0; jy < 6; ++jy) {
        int base = iy[jy] << 10;
        float acc = 0.0f;
#pragma unroll
        for (int jx = 0; jx < 6; ++jx)
            acc += grid_r[base + ix[jx]] * wx[jx];
        v += acc * wy[jy];
    }
    w[k] = w[k] / fmaxf(fabsf(v), 1e-12f);
}

// Final complex gridding of kd * dcomp.
__global__ __launch_bounds__(128) void scatter_cplx(const float* __restrict__ ksp,
                                                    const float* __restrict__ kd_r,
                                                    const float* __restrict__ kd_i,
                                                    const float* __restrict__ w,
                                                    float* __restrict__ grid_r,
                                                    float* __restrict__ grid_i) {
    int k = blockIdx.x * blockDim.x + threadIdx.x;
    if (k >= KPTS) return;
    float inv_i0a = 1.0f / kb_i0f(ALPHA_KB);
    float wy[6]; int iy[6];
    float wx[6]; int ix[6];
    kb6(ksp[k], wy, iy, inv_i0a);
    kb6(ksp[KPTS + k], wx, ix, inv_i0a);
    float dr = kd_r[k] * w[k];
    float di = kd_i[k] * w[k];
#pragma unroll
    for (int jy = 0; jy < 6; ++jy) {
        int base = iy[jy] << 10;
        float wyj = wy[jy];
#pragma unroll
        for (int jx = 0; jx < 6; ++jx) {
            float ww = wyj * wx[jx];
            atomicAdd(&grid_r[base + ix[jx]], dr * ww);
            atomicAdd(&grid_i[base + ix[jx]], di * ww);
        }
    }
}

// ---------------------------------------------------------------------------
// 1024-pt inverse FFT via four-step (32x32) with V_WMMA_F32_16X16X4_F32.
//   x[n1 + 32*n2] -> M[n1][n2];  Y = M * F;  Z = Y .* W1024^{+n1*k2};  R = F * Z
//   R[k1][k2] = X[32*k1 + k2].  F[n][k] = e^{+2*pi*i*n*k/32} (symmetric).
// One FFT per 128-thread workgroup (4 wave32 tiles of 16x16), output stored
// TRANSPOSED so two passes give the full 2D inverse transform in natural order.
// ---------------------------------------------------------------------------
__global__ __launch_bounds__(128) void ifft1024_pass(const float* __restrict__ in_r,
                                                     const float* __restrict__ in_i,
                                                     float* __restrict__ out_r,
                                                     float* __restrict__ out_i) {
    __shared__ float Mr[32][33], Mi[32][33];
    __shared__ float Zr[32][33], Zi[32][33];
    __shared__ float Fr[32][33], Fi[32][33];
    const int row = blockIdx.x;
    const int tid = threadIdx.x;

    for (int e = tid; e < 1024; e += 128) {
        int n1 = e & 31, n2 = e >> 5;
        Mr[n1][n2] = in_r[row * 1024 + e];
        Mi[n1][n2] = in_i[row * 1024 + e];
        int fn = e >> 5, fk = e & 31;
        float ang = (2.0f * PI_F / 32.0f) * (float)((fn * fk) & 31);
        Fr[fn][fk] = cosf(ang);
        Fi[fn][fk] = sinf(ang);
    }
    __syncthreads();

    const int lane = tid & 31;
    const int wv = tid >> 5;        // wave 0..3
    const int ti = (wv >> 1) & 1;   // 16-row tile
    const int tj = wv & 1;          // 16-col tile
    const int h = lane >> 4;        // lane half (K interleave)
    const int lm = lane & 15;

    // ---- Stage A: Y = M * F, twiddle, -> Z ----
    {
        v8f cr = {0.f,0.f,0.f,0.f,0.f,0.f,0.f,0.f};
        v8f ci = {0.f,0.f,0.f,0.f,0.f,0.f,0.f,0.f};
        const int am = ti * 16 + lm;
        const int bn = tj * 16 + lm;
#pragma unroll
        for (int kk = 0; kk < 8; ++kk) {
            const int ka = kk * 4 + 2 * h;   // A/B: VGPR v, half h -> K = v + 2h
            v2f ar  = {  Mr[am][ka],  Mr[am][ka + 1] };
            v2f ain = { -Mi[am][ka], -Mi[am][ka + 1] };
            v2f aip = {  Mi[am][ka],  Mi[am][ka + 1] };
            v2f br  = {  Fr[ka][bn],  Fr[ka + 1][bn] };
            v2f bi  = {  Fi[ka][bn],  Fi[ka + 1][bn] };
            cr = __builtin_amdgcn_wmma_f32_16x16x4_f32(false, ar,  false, br, (short)0, cr, false, false);
            cr = __builtin_amdgcn_wmma_f32_16x16x4_f32(false, ain, false, bi, (short)0, cr, false, false);
            ci = __builtin_amdgcn_wmma_f32_16x16x4_f32(false, ar,  false, bi, (short)0, ci, false, false);
            ci = __builtin_amdgcn_wmma_f32_16x16x4_f32(false, aip, false, br, (short)0, ci, false, false);
        }
#pragma unroll
        for (int i = 0; i < 8; ++i) {
            int m = ti * 16 + i + 8 * h;     // n1  (C/D: VGPR i -> M = i + 8*half)
            int n = tj * 16 + lm;            // k2
            float ang = (2.0f * PI_F / 1024.0f) * (float)(m * n);
            float cs = cosf(ang), sn = sinf(ang);
            float yr = cr[i], yi = ci[i];
            Zr[m][n] = yr * cs - yi * sn;
            Zi[m][n] = yr * sn + yi * cs;
        }
    }
    __syncthreads();

    // ---- Stage B: R = F * Z, store transposed ----
    {
        v8f cr = {0.f,0.f,0.f,0.f,0.f,0.f,0.f,0.f};
        v8f ci = {0.f,0.f,0.f,0.f,0.f,0.f,0.f,0.f};
        const int am = ti * 16 + lm;         // k1 (F symmetric)
        const int bn = tj * 16 + lm;         // k2
#pragma unroll
        for (int kk = 0; kk < 8; ++kk) {
            const int ka = kk * 4 + 2 * h;
            v2f ar  = {  Fr[am][ka],  Fr[am][ka + 1] };
            v2f ain = { -Fi[am][ka], -Fi[am][ka + 1] };
            v2f aip = {  Fi[am][ka],  Fi[am][ka + 1] };
            v2f br  = {  Zr[ka][bn],  Zr[ka + 1][bn] };
            v2f bi  = {  Zi[ka][bn],  Zi[ka + 1][bn] };
            cr = __builtin_amdgcn_wmma_f32_16x16x4_f32(false, ar,  false, br, (short)0, cr, false, false);
            cr = __builtin_amdgcn_wmma_f32_16x16x4_f32(false, ain, false, bi, (short)0, cr, false, false);
            ci = __builtin_amdgcn_wmma_f32_16x16x4_f32(false, ar,  false, bi, (short)0, ci, false, false);
            ci = __builtin_amdgcn_wmma_f32_16x16x4_f32(false, aip, false, br, (short)0, ci, false, false);
        }
#pragma unroll
        for (int i = 0; i < 8; ++i) {
            int k1 = ti * 16 + i + 8 * h;
            int k2 = tj * 16 + lm;
            int q = (k1 << 5) + k2;          // output index within the FFT
            out_r[(size_t)q * 1024 + row] = cr[i];  // transposed store
            out_i[(size_t)q * 1024 + row] = ci[i];
        }
    }
}

// ---------------------------------------------------------------------------
// fftshift + center crop + deapodization + ortho scale (1/1024).
// ---------------------------------------------------------------------------
__device__ __forceinline__ float deapod512(int p, float inv_i0a) {
    float n = (float)p - 256.0f;
    float xi = n / 1024.0f;
    float u = PI_F * 6.0f * xi;
    float t = u * u - ALPHA_KB * ALPHA_KB;
    float z = sqrtf(fabsf(t));
    float kbft;
    if (t < 0.0f) kbft = sinhf(z) / fmaxf(z, 1e-12f);
    else          kbft = (z > 1e-12f) ? (sinf(z) / z) : 1.0f;
    kbft *= 6.0f * inv_i0a;
    return 1.0f / kbft;
}

__global__ void crop_deapod(const float* __restrict__ img_r, float* __restrict__ out) {
    int i = blockIdx.x * blockDim.x + threadIdx.x;
    if (i >= 512 * 512) return;
    int y = i >> 9, x = i & 511;
    int sr = (y + 768) & 1023;               // fftshift + crop fold
    int sc = (x + 768) & 1023;
    float inv_i0a = 1.0f / kb_i0f(ALPHA_KB);
    float v = img_r[sr * 1024 + sc];
    out[i] = v * deapod512(y, inv_i0a) * deapod512(x, inv_i0a) * (1.0f / 1024.0f);
}

// ---------------------------------------------------------------------------
// Host-side orchestration (graph-capture safe: only kernel launches on stream).
// ---------------------------------------------------------------------------
extern "C" void kernel_launch(void* const* d_in, const int* in_sizes, int n_in,
                              void* d_out, int out_size, void* d_ws, size_t ws_size,
                              hipStream_t stream) {
    (void)in_sizes; (void)n_in; (void)out_size; (void)ws_size;
    const float* sino = (const float*)d_in[0];   // (720, 512) f32
    const float* ksp  = (const float*)d_in[1];   // (2, K) f32
    float* out = (float*)d_out;                  // (1,1,512,512) f32

    char* ws = (char*)d_ws;
    const size_t GRIDB = (size_t)GSZ * GSZ * sizeof(float);  // 4 MB
    const size_t KB4 = (size_t)KPTS * sizeof(float);
    float* grid_r = (float*)(ws);
    float* grid_i = (float*)(ws + GRIDB);        // contiguous with grid_r
    float* T_r    = (float*)(ws + 2 * GRIDB);
    float* T_i    = (float*)(ws + 3 * GRIDB);
    float* kd_r   = (float*)(ws + 4 * GRIDB);
    float* kd_i   = (float*)(ws + 4 * GRIDB + KB4);
    float* w      = (float*)(ws + 4 * GRIDB + 2 * KB4);

    const int gcells = GSZ * GSZ;

    // 1) sinogram -> k-space samples (row FFTs with folded shifts)
    fft512_rows<<<NANG, 256, 0, stream>>>(sino, kd_r, kd_i);

    // 2) Pipe-Menon density compensation (purely real)
    fill_f32<<<(KPTS + 255) / 256, 256, 0, stream>>>(w, 1.0f, KPTS);
    for (int it = 0; it < NITER_DCOMP; ++it) {
        zero_f32<<<(gcells + 255) / 256, 256, 0, stream>>>(grid_r, gcells);
        scatter_real<<<(KPTS + 127) / 128, 128, 0, stream>>>(ksp, w, grid_r);
        gather_update<<<(KPTS + 127) / 128, 128, 0, stream>>>(ksp, grid_r, w);
    }

    // 3) final complex gridding of kd * dcomp
    zero_f32<<<(2 * gcells + 255) / 256, 256, 0, stream>>>(grid_r, 2 * gcells);
    scatter_cplx<<<(KPTS + 127) / 128, 128, 0, stream>>>(ksp, kd_r, kd_i, w, grid_r, grid_i);

    // 4) 1024x1024 ortho IFFT2 as two transposing WMMA four-step passes
    ifft1024_pass<<<GSZ, 128, 0, stream>>>(grid_r, grid_i, T_r, T_i);
    ifft1024_pass<<<GSZ, 128, 0, stream>>>(T_r, T_i, grid_r, grid_i);

    // 5) fftshift + crop + deapodize + 1/1024 scale
    crop_deapod<<<(512 * 512 + 255) / 256, 256, 0, stream>>>(grid_r, out);
}